// GATE_29910152249807
// MI455X (gfx1250) — compile-verified
//
#include <hip/hip_runtime.h>
#include <hip/hip_bf16.h>

typedef __attribute__((ext_vector_type(2))) float v2f;
typedef __attribute__((ext_vector_type(8))) float v8f;

// =====================================================================
// GEMM: C[n,DOUT] = A[n,DIN] @ op(B) (+ bias), fp32 via V_WMMA_F32_16X16X4_F32
//   TRANSB=true : B is [DOUT,DIN] row-major, op(B)=B^T   (encoder H @ W^T)
//   TRANSB=false: B is [DIN,DOUT] row-major, op(B)=B     (decoder H @ W)
// One wave -> one 16x16 C tile. 8 waves/block share a 16-row A strip in LDS
// (padded stride to avoid bank conflicts). W matrices are tiny + L2 resident.
// =====================================================================
template <int DIN, int DOUT, bool TRANSB, bool BIAS>
__global__ __launch_bounds__(256) void gemm_wmma_f32(
    const float* __restrict__ A, const float* __restrict__ B,
    const float* __restrict__ bias, float* __restrict__ C)
{
  constexpr int LDA = DIN + 4;              // pad: lanes land on distinct banks
  __shared__ float As[16 * LDA];

  const int row0 = blockIdx.x * 16;
  const int wave = threadIdx.x >> 5;
  const int lane = threadIdx.x & 31;
  const int j0   = (blockIdx.y * 8 + wave) * 16;

  // cooperative, fully-coalesced load of the contiguous 16 x DIN A strip
  for (int idx = threadIdx.x; idx < 16 * DIN; idx += 256) {
    int r = idx / DIN, c = idx - r * DIN;
    As[r * LDA + c] = A[(size_t)row0 * DIN + idx];
  }
  __syncthreads();

  const int m  = lane & 15;           // A row within tile
  const int nn = lane & 15;           // B/C column within tile
  const int kp = (lane >> 4) << 1;    // K pair owned by this half-wave

  v8f acc = {};
  #pragma unroll 4
  for (int k0 = 0; k0 < DIN; k0 += 4) {
    v2f a, b;
    a.x = As[m * LDA + k0 + kp];
    a.y = As[m * LDA + k0 + kp + 1];
    if (TRANSB) {
      b.x = B[(size_t)(j0 + nn) * DIN + k0 + kp];
      b.y = B[(size_t)(j0 + nn) * DIN + k0 + kp + 1];
    } else {
      b.x = B[(size_t)(k0 + kp) * DOUT + j0 + nn];
      b.y = B[(size_t)(k0 + kp + 1) * DOUT + j0 + nn];
    }
    acc = __builtin_amdgcn_wmma_f32_16x16x4_f32(false, a, false, b,
                                                (short)0, acc, false, false);
  }

  const float bj = BIAS ? bias[j0 + nn] : 0.0f;
  #pragma unroll
  for (int r = 0; r < 8; ++r) {
    int mm = r + ((lane >> 4) << 3);  // C layout: VGPR r -> rows r / r+8
    C[(size_t)(row0 + mm) * DOUT + j0 + nn] = acc[r] + bj;
  }
}

// ---------------- per-node attention logits: f1 = M@va, f2 = M@vb ----------
template <int D>
__global__ __launch_bounds__(256) void node_logits(
    const float* __restrict__ M, const float* __restrict__ va,
    const float* __restrict__ vb, float* __restrict__ f1,
    float* __restrict__ f2, int n)
{
  int node = blockIdx.x * 8 + (threadIdx.x >> 5);
  if (node >= n) return;
  int lane = threadIdx.x & 31;
  float s1 = 0.f, s2 = 0.f;
  #pragma unroll
  for (int c = lane; c < D; c += 32) {
    float h = M[(size_t)node * D + c];
    s1 += h * va[c];
    s2 += h * vb[c];
  }
  #pragma unroll
  for (int off = 16; off; off >>= 1) {
    s1 += __shfl_xor(s1, off, 32);
    s2 += __shfl_xor(s2, off, 32);
  }
  if (lane == 0) { f1[node] = s1; f2[node] = s2; }
}

// ---------------- misc ------------------------------------------------------
__global__ __launch_bounds__(256) void fill_zero(float* __restrict__ p, size_t n)
{
  size_t i = (size_t)blockIdx.x * blockDim.x + threadIdx.x;
  if (i < n) p[i] = 0.0f;
}

// ---------------- edge softmax (segment by source row) ----------------------
// vals = sigmoid(A_vals*(f1[row]+f2[col])) in (0,1): positive-float bit
// pattern is monotone, so segment-max is a uint atomicMax with identity 0.
__global__ __launch_bounds__(256) void edge_sig_max(
    const int* __restrict__ rows, const int* __restrict__ cols,
    const float* __restrict__ av, const float* __restrict__ f1,
    const float* __restrict__ f2, float* __restrict__ att,
    unsigned* __restrict__ mbits, int E)
{
  int i = blockIdx.x * blockDim.x + threadIdx.x;
  if (i >= E) return;
  int r = rows[i];
  float z = av[i] * (f1[r] + f2[cols[i]]);
  float val = 1.0f / (1.0f + expf(-z));
  att[i] = val;
  atomicMax(&mbits[r], __float_as_uint(val));
}

__global__ __launch_bounds__(256) void edge_exp_sum(
    const int* __restrict__ rows, float* __restrict__ att,
    const unsigned* __restrict__ mbits, float* __restrict__ s, int E)
{
  int i = blockIdx.x * blockDim.x + threadIdx.x;
  if (i >= E) return;
  int r = rows[i];
  float e = expf(att[i] - __uint_as_float(mbits[r]));
  att[i] = e;
  atomicAdd(&s[r], e);
}

__global__ __launch_bounds__(256) void edge_norm(
    const int* __restrict__ rows, float* __restrict__ att,
    const float* __restrict__ s, int E)
{
  int i = blockIdx.x * blockDim.x + threadIdx.x;
  if (i >= E) return;
  att[i] = att[i] / s[rows[i]];
}

// ---------------- SpMM: out[row] += att[e] * H[col], edge-parallel ----------
// H rows (2KB max) are contiguous and L2-resident (<=102MB << 192MB L2).
// D4 = feature_dim/4; one thread handles one float4 of one edge.
template <int D4>
__global__ __launch_bounds__(256) void spmm_scatter(
    const int* __restrict__ rows, const int* __restrict__ cols,
    const float* __restrict__ att, const float* __restrict__ H,
    float* __restrict__ out, long long total)
{
  long long idx = (long long)blockIdx.x * blockDim.x + threadIdx.x;
  if (idx >= total) return;
  int e = (int)(idx / D4);
  int c = (int)(idx - (long long)e * D4);
  int r = rows[e], co = cols[e];
  float a = att[e];
  const float4 v = reinterpret_cast<const float4*>(H)[(size_t)co * D4 + c];
  float* o = out + ((size_t)r * D4 + c) * 4;
  atomicAdd(o + 0, a * v.x);
  atomicAdd(o + 1, a * v.y);
  atomicAdd(o + 2, a * v.z);
  atomicAdd(o + 3, a * v.w);
}

// =====================================================================
extern "C" void kernel_launch(void* const* d_in, const int* in_sizes, int n_in,
                              void* d_out, int out_size, void* d_ws, size_t ws_size,
                              hipStream_t stream)
{
  const float* X   = (const float*)d_in[0];
  const int*   erow = (const int*)d_in[1];
  const int*   ecol = (const int*)d_in[2];
  const float* av  = (const float*)d_in[3];
  const float* W0  = (const float*)d_in[4];   // [256,512]
  const float* b0  = (const float*)d_in[5];
  const float* W1  = (const float*)d_in[6];   // [128,256]
  const float* b1  = (const float*)d_in[7];
  const float* v00 = (const float*)d_in[8];
  const float* v01 = (const float*)d_in[9];
  const float* v10 = (const float*)d_in[10];
  const float* v11 = (const float*)d_in[11];

  const int N = in_sizes[0] / 512;            // 50000 (divisible by 16)
  const int E = in_sizes[1];                  // 1.6M

  float* ws   = (float*)d_ws;                 // ~244 MB total
  float* buf1 = ws;                           // N*256  (M0, later D1)
  float* buf2 = buf1 + (size_t)N * 256;       // N*256  (H1, later D2)
  float* buf3 = buf2 + (size_t)N * 256;       // N*128  (M1)
  float* buf4 = buf3 + (size_t)N * 128;       // N*512  (D3)
  float* att0 = buf4 + (size_t)N * 512;       // E
  float* att1 = att0 + (size_t)E;             // E
  float* f1   = att1 + (size_t)E;             // N
  float* f2   = f1 + N;                       // N
  float* mseg = f2 + N;                       // N (uint max bits)
  float* sseg = mseg + N;                     // N (contiguous with mseg)

  float* outH  = (float*)d_out;                    // [N,512]
  float* outFH = (float*)d_out + (size_t)N * 512;  // [N,128] final_H (= H2)

  const int rowTiles = N / 16;
  dim3 blk(256);
  dim3 edgeGrid((E + 255) / 256);
  dim3 nodeGrid((N + 7) / 8);
  auto g1 = [](size_t n) { return dim3((unsigned)((n + 255) / 256)); };

  // ---------------- encoder layer 0 ----------------
  gemm_wmma_f32<512, 256, true, true><<<dim3(rowTiles, 2), blk, 0, stream>>>(X, W0, b0, buf1);
  node_logits<256><<<nodeGrid, blk, 0, stream>>>(buf1, v00, v01, f1, f2, N);
  fill_zero<<<g1((size_t)2 * N), blk, 0, stream>>>(mseg, (size_t)2 * N);
  edge_sig_max<<<edgeGrid, blk, 0, stream>>>(erow, ecol, av, f1, f2, att0, (unsigned*)mseg, E);
  edge_exp_sum<<<edgeGrid, blk, 0, stream>>>(erow, att0, (const unsigned*)mseg, sseg, E);
  edge_norm<<<edgeGrid, blk, 0, stream>>>(erow, att0, sseg, E);
  fill_zero<<<g1((size_t)N * 256), blk, 0, stream>>>(buf2, (size_t)N * 256);
  spmm_scatter<64><<<g1((size_t)E * 64), blk, 0, stream>>>(erow, ecol, att0, buf1, buf2, (long long)E * 64);

  // ---------------- encoder layer 1 ----------------
  gemm_wmma_f32<256, 128, true, true><<<dim3(rowTiles, 1), blk, 0, stream>>>(buf2, W1, b1, buf3);
  node_logits<128><<<nodeGrid, blk, 0, stream>>>(buf3, v10, v11, f1, f2, N);
  fill_zero<<<g1((size_t)2 * N), blk, 0, stream>>>(mseg, (size_t)2 * N);
  edge_sig_max<<<edgeGrid, blk, 0, stream>>>(erow, ecol, av, f1, f2, att1, (unsigned*)mseg, E);
  edge_exp_sum<<<edgeGrid, blk, 0, stream>>>(erow, att1, (const unsigned*)mseg, sseg, E);
  edge_norm<<<edgeGrid, blk, 0, stream>>>(erow, att1, sseg, E);
  fill_zero<<<g1((size_t)N * 128), blk, 0, stream>>>(outFH, (size_t)N * 128);
  spmm_scatter<32><<<g1((size_t)E * 32), blk, 0, stream>>>(erow, ecol, att1, buf3, outFH, (long long)E * 32);

  // ---------------- decoder (tied weights, reused attention) ----------------
  gemm_wmma_f32<128, 256, false, false><<<dim3(rowTiles, 2), blk, 0, stream>>>(outFH, W1, nullptr, buf1);
  fill_zero<<<g1((size_t)N * 256), blk, 0, stream>>>(buf2, (size_t)N * 256);
  spmm_scatter<64><<<g1((size_t)E * 64), blk, 0, stream>>>(erow, ecol, att1, buf1, buf2, (long long)E * 64);
  gemm_wmma_f32<256, 512, false, false><<<dim3(rowTiles, 4), blk, 0, stream>>>(buf2, W0, nullptr, buf4);
  fill_zero<<<g1((size_t)N * 512), blk, 0, stream>>>(outH, (size_t)N * 512);
  spmm_scatter<128><<<g1((size_t)E * 128), blk, 0, stream>>>(erow, ecol, att0, buf4, outH, (long long)E * 128);
}